// YOLOv4_40750649704848
// MI455X (gfx1250) — compile-verified
//
#include <hip/hip_runtime.h>
#include <stdint.h>

#define C_CLS 20
#define A_ANCH 3
#define HIST_BINS 4096
#define CAND_CAP 4096

// ---------------- ws layout (bytes) ----------------
#define WS_HIST      0          // 3*4096*4 = 49152
#define WS_COUNTERS  49152      // 3*4
#define WS_THR       49168      // 3*4
#define WS_CAND      49664      // 3*4096*8 = 98304
#define WS_TOPIDX    147968     // 3*128*4
#define WS_TOPSCORE  149504     // 3*128*4  (total ~151040 B)

__device__ __constant__ float c_aw[3][3] = {{12.f,19.f,40.f},{36.f,76.f,72.f},{142.f,192.f,459.f}};
__device__ __constant__ float c_ah[3][3] = {{16.f,36.f,28.f},{75.f,55.f,146.f},{110.f,243.f,401.f}};

__device__ __forceinline__ float sigf(float x) { return 1.0f / (1.0f + expf(-x)); }

// ---------------- init: zero histograms + counters ----------------
__global__ void init_kernel(unsigned* __restrict__ hist, unsigned* __restrict__ counters) {
  int i = blockIdx.x * blockDim.x + threadIdx.x;
  if (i < 3 * HIST_BINS) hist[i] = 0u;
  if (i < 3) counters[i] = 0u;
}

// ---------------- pass 1: score histogram (async LDS double-buffer stream) ----------------
__global__ void hist_kernel(const float* __restrict__ obj, const float* __restrict__ cls,
                            int HW, unsigned* __restrict__ ghist, int nchunks) {
  __shared__ unsigned hist[HIST_BINS];
  __shared__ float4 stage[2][256];
  const int tid = threadIdx.x;
  for (int b = tid; b < HIST_BINS; b += 256) hist[b] = 0u;
  __syncthreads();

  // LDS byte offset of this lane's slot (low 32 bits of generic LDS pointer)
  const unsigned lane_off = (unsigned)(uintptr_t)(&stage[0][0]) + (unsigned)tid * 16u;

  int ch = blockIdx.x;
  if (ch < nchunks) {
    // preload first chunk into buffer 0 via CDNA5 async global->LDS DMA path
    unsigned long long g0 =
        (unsigned long long)(uintptr_t)cls + ((unsigned long long)ch * 256u + (unsigned)tid) * 16ull;
    asm volatile("global_load_async_to_lds_b128 %0, %1, off" :: "v"(lane_off), "v"(g0) : "memory");
    int buf = 0;
    while (ch < nchunks) {
      int next = ch + (int)gridDim.x;
      if (next < nchunks) {
        unsigned long long gn =
            (unsigned long long)(uintptr_t)cls + ((unsigned long long)next * 256u + (unsigned)tid) * 16ull;
        unsigned dstoff = lane_off + (buf ? 0u : 4096u); // other buffer
        asm volatile("global_load_async_to_lds_b128 %0, %1, off" :: "v"(dstoff), "v"(gn) : "memory");
        asm volatile("s_wait_asynccnt 0x1" ::: "memory"); // current chunk landed (in-order)
      } else {
        asm volatile("s_wait_asynccnt 0x0" ::: "memory");
      }
      float4 cv = stage[buf][tid];                 // each lane reads only what it DMA'd
      unsigned e = ((unsigned)ch * 256u + (unsigned)tid) * 4u;
      unsigned plane = e / (unsigned)HW;           // HW % 4 == 0 -> whole float4 in one plane
      unsigned pos   = e - plane * (unsigned)HW;
      unsigned a     = plane / C_CLS;
      const float4 ov = *(const float4*)(obj + (size_t)a * (size_t)HW + pos);
      float so[4] = { sigf(ov.x), sigf(ov.y), sigf(ov.z), sigf(ov.w) };
      float sc[4] = { sigf(cv.x), sigf(cv.y), sigf(cv.z), sigf(cv.w) };
#pragma unroll
      for (int k = 0; k < 4; ++k) {
        unsigned bbits = __float_as_uint(so[k] * sc[k]);
        if (bbits >= 0x3F000000u) {               // score >= 0.5
          unsigned bin = (bbits - 0x3F000000u) >> 11;
          if (bin > HIST_BINS - 1u) bin = HIST_BINS - 1u;
          atomicAdd(&hist[bin], 1u);
        }
      }
      ch = next; buf ^= 1;
    }
  }
  __syncthreads();
  for (int b = tid; b < HIST_BINS; b += 256) {
    unsigned v = hist[b];
    if (v) atomicAdd(&ghist[b], v);
  }
}

// ---------------- pass 2: find per-level score threshold (top-100 cut) ----------------
__global__ void thresh_kernel(const unsigned* __restrict__ ghist, unsigned* __restrict__ thr) {
  const int lvl = blockIdx.x;
  const unsigned* h = ghist + lvl * HIST_BINS;
  const int tid = threadIdx.x; // 256 threads, 16 bins each
  __shared__ unsigned chunk[256];
  __shared__ unsigned suffix[257];
  __shared__ int redmax;
  unsigned loc[16]; unsigned s = 0u;
#pragma unroll
  for (int i = 0; i < 16; ++i) { loc[i] = h[tid * 16 + i]; s += loc[i]; }
  chunk[tid] = s;
  if (tid == 0) redmax = -1;
  __syncthreads();
  if (tid == 0) { suffix[256] = 0u; for (int i = 255; i >= 0; --i) suffix[i] = suffix[i + 1] + chunk[i]; }
  __syncthreads();
  int best = -1;
  unsigned run = suffix[tid + 1];
  for (int i = 15; i >= 0; --i) { run += loc[i]; if (run >= 100u) { best = tid * 16 + i; break; } }
  atomicMax(&redmax, best);
  __syncthreads();
  if (tid == 0) thr[lvl] = (redmax < 0) ? 0u : (0x3F000000u + ((unsigned)redmax << 11));
}

// ---------------- pass 3: collect candidates >= threshold ----------------
__global__ void collect_kernel(const float* __restrict__ obj, const float* __restrict__ cls,
                               int HW, const unsigned* __restrict__ thr, int lvl,
                               unsigned* __restrict__ counter,
                               unsigned long long* __restrict__ cand, int E4) {
  const unsigned tbits = thr[lvl];
  const int step = gridDim.x * blockDim.x;
  for (int e4 = blockIdx.x * blockDim.x + threadIdx.x; e4 < E4; e4 += step) {
    // speculative prefetch of next grid-stride iteration (global_prefetch_b8)
    __builtin_prefetch(&((const float4*)cls)[e4 + step], 0, 1);
    unsigned e = (unsigned)e4 * 4u;
    unsigned plane = e / (unsigned)HW;
    unsigned pos   = e - plane * (unsigned)HW;
    unsigned a     = plane / C_CLS;
    unsigned c     = plane - a * C_CLS;
    float4 cv = ((const float4*)cls)[e4];
    float4 ov = *(const float4*)(obj + (size_t)a * (size_t)HW + pos);
    float so[4] = { sigf(ov.x), sigf(ov.y), sigf(ov.z), sigf(ov.w) };
    float sc[4] = { sigf(cv.x), sigf(cv.y), sigf(cv.z), sigf(cv.w) };
#pragma unroll
    for (int k = 0; k < 4; ++k) {
      unsigned bbits = __float_as_uint(so[k] * sc[k]);
      if (bbits >= tbits) {
        unsigned p = atomicAdd(counter, 1u);
        if (p < CAND_CAP) {
          unsigned flat = ((pos + (unsigned)k) * A_ANCH + a) * C_CLS + c; // == top_i of reference
          cand[p] = ((unsigned long long)bbits << 32) | (unsigned long long)(0xFFFFFFFFu - flat);
        }
      }
    }
  }
}

// ---------------- pass 4: per-level bitonic sort, emit top-100 ----------------
__global__ void sortlevel_kernel(const unsigned* __restrict__ counters,
                                 const unsigned long long* __restrict__ cand_all,
                                 unsigned* __restrict__ top_idx, float* __restrict__ top_score) {
  const int lvl = blockIdx.x;
  const unsigned long long* cand = cand_all + (size_t)lvl * CAND_CAP;
  __shared__ unsigned long long key[CAND_CAP];
  const int tid = threadIdx.x;
  unsigned n = counters[lvl]; if (n > CAND_CAP) n = CAND_CAP;
  for (int i = tid; i < CAND_CAP; i += 256) key[i] = (i < (int)n) ? cand[i] : 0ull;
  __syncthreads();
  for (int k = 2; k <= CAND_CAP; k <<= 1)
    for (int j = k >> 1; j > 0; j >>= 1) {
      for (int i = tid; i < CAND_CAP; i += 256) {
        int ixj = i ^ j;
        if (ixj > i) {
          unsigned long long x = key[i], y = key[ixj];
          bool up = (i & k) == 0;                  // descending overall
          if (up ? (x < y) : (x > y)) { key[i] = y; key[ixj] = x; }
        }
      }
      __syncthreads();
    }
  if (tid < 100) {
    unsigned long long kk = key[tid];
    if (kk == 0ull) { top_score[lvl * 128 + tid] = 0.f; top_idx[lvl * 128 + tid] = 0u; }
    else {
      top_score[lvl * 128 + tid] = __uint_as_float((unsigned)(kk >> 32));
      top_idx[lvl * 128 + tid]   = 0xFFFFFFFFu - (unsigned)kk;
    }
  }
}

// ---------------- pass 5: decode, global sort, matrix NMS, write output ----------------
__global__ void final_kernel(const unsigned* __restrict__ top_idx, const float* __restrict__ top_score,
                             const float* __restrict__ reg0, const float* __restrict__ reg1,
                             const float* __restrict__ reg2, float* __restrict__ out) {
  __shared__ unsigned long long key[512];
  __shared__ float b0[300], b1[300], b2[300], b3[300];
  __shared__ int   lab[300];
  __shared__ float sb0[300], sb1[300], sb2[300], sb3[300], ss[300], sarea[300];
  __shared__ int   sl[300], sval[300], skeep[300];
  __shared__ unsigned supmat[300][10];   // bit j of row i: i suppresses j (j>i, same label, iou>0.5)
  __shared__ unsigned supvec[10];        // running suppressed bitmask
  const int tid = threadIdx.x;

  if (tid < 300) {
    int lvl = tid / 100, r = tid - lvl * 100;
    float s = top_score[lvl * 128 + r];
    unsigned f = top_idx[lvl * 128 + r];
    int c = (int)(f % C_CLS);
    unsigned n = f / C_CLS;
    int a = (int)(n % A_ANCH);
    unsigned pos = n / A_ANCH;
    int W = (lvl == 0) ? 320 : ((lvl == 1) ? 160 : 80);
    int HW = W * W;
    float stride = (lvl == 0) ? 8.f : ((lvl == 1) ? 16.f : 32.f);
    const float* reg = (lvl == 0) ? reg0 : ((lvl == 1) ? reg1 : reg2);
    int x = (int)(pos % (unsigned)W), y = (int)(pos / (unsigned)W);
    float tx = reg[(size_t)(a * 4 + 0) * HW + pos];
    float ty = reg[(size_t)(a * 4 + 1) * HW + pos];
    float tw = reg[(size_t)(a * 4 + 2) * HW + pos];
    float th = reg[(size_t)(a * 4 + 3) * HW + pos];
    float cx = (sigf(tx) * 3.f - 1.5f + (float)x + 0.5f) * stride;
    float cy = (sigf(ty) * 3.f - 1.5f + (float)y + 0.5f) * stride;
    float ww = expf(tw) * c_aw[lvl][a];
    float hh = expf(th) * c_ah[lvl][a];
    b0[tid] = cx - 0.5f * ww; b1[tid] = cy - 0.5f * hh;
    b2[tid] = cx + 0.5f * ww; b3[tid] = cy + 0.5f * hh;
    lab[tid] = c;
    key[tid] = ((unsigned long long)__float_as_uint(s) << 32) |
               (unsigned long long)(0xFFFFFFFFu - (unsigned)tid); // stable argsort(-s)
  } else {
    key[tid] = 0ull;
  }
  for (int w = tid; w < 3000; w += 512) ((unsigned*)supmat)[w] = 0u;
  if (tid < 10) supvec[tid] = 0u;
  __syncthreads();

  // bitonic sort 512 keys, descending
  for (int k = 2; k <= 512; k <<= 1)
    for (int j = k >> 1; j > 0; j >>= 1) {
      int i = tid, ixj = i ^ j;
      if (ixj > i) {
        unsigned long long x = key[i], y = key[ixj];
        bool up = (i & k) == 0;
        if (up ? (x < y) : (x > y)) { key[i] = y; key[ixj] = x; }
      }
      __syncthreads();
    }

  // gather into sorted order
  if (tid < 300) {
    unsigned long long kk = key[tid];
    int orig = (int)(0xFFFFFFFFu - (unsigned)kk); // always < 300 (all real scores > 0)
    float s = __uint_as_float((unsigned)(kk >> 32));
    sb0[tid] = b0[orig]; sb1[tid] = b1[orig]; sb2[tid] = b2[orig]; sb3[tid] = b3[orig];
    ss[tid] = s; sl[tid] = lab[orig];
    sval[tid] = (s > 0.01f) ? 1 : 0;
    sarea[tid] = (sb2[tid] - sb0[tid]) * (sb3[tid] - sb1[tid]);
    skeep[tid] = 0;
  }
  __syncthreads();

  // build full suppression matrix in parallel (45k useful pairs / 512 threads)
  for (int idx = tid; idx < 300 * 300; idx += 512) {
    int i = idx / 300, j = idx - i * 300;
    if (j > i && sl[i] == sl[j]) {
      float xx1 = fmaxf(sb0[i], sb0[j]), yy1 = fmaxf(sb1[i], sb1[j]);
      float xx2 = fminf(sb2[i], sb2[j]), yy2 = fminf(sb3[i], sb3[j]);
      float inter = fmaxf(1e-10f, xx2 - xx1) * fmaxf(1e-10f, yy2 - yy1);
      float iou = inter / (sarea[i] + sarea[j] - inter);
      if (iou > 0.5f) atomicOr(&supmat[i][j >> 5], 1u << (j & 31));
    }
  }
  __syncthreads();

  // greedy scan: single wave32, no barriers (wave-internal LDS ops are in order)
  if (tid < 32) {
    for (int i = 0; i < 300; ++i) {
      unsigned w = supvec[i >> 5];                      // same address: broadcast to all lanes
      bool is_kept = (((w >> (i & 31)) & 1u) == 0u) && (sval[i] != 0);
      if (tid == 0) skeep[i] = is_kept ? 1 : 0;
      if (is_kept && tid < 10) supvec[tid] |= supmat[i][tid];
    }
  }
  __syncthreads();

  for (int t = tid; t < 300; t += blockDim.x) {
    out[t * 4 + 0] = sb0[t]; out[t * 4 + 1] = sb1[t];
    out[t * 4 + 2] = sb2[t]; out[t * 4 + 3] = sb3[t];
    out[1200 + t] = ss[t] * (float)skeep[t];
    out[1500 + t] = (float)sl[t];
    out[1800 + t] = (float)skeep[t];
  }
}

// ---------------- host launcher ----------------
extern "C" void kernel_launch(void* const* d_in, const int* in_sizes, int n_in,
                              void* d_out, int out_size, void* d_ws, size_t ws_size,
                              hipStream_t stream) {
  (void)in_sizes; (void)n_in; (void)out_size; (void)ws_size;
  const float* obj[3] = { (const float*)d_in[0], (const float*)d_in[3], (const float*)d_in[6] };
  const float* cls[3] = { (const float*)d_in[1], (const float*)d_in[4], (const float*)d_in[7] };
  const float* reg[3] = { (const float*)d_in[2], (const float*)d_in[5], (const float*)d_in[8] };
  const int HWs[3] = { 320 * 320, 160 * 160, 80 * 80 };

  uint8_t* ws = (uint8_t*)d_ws;
  unsigned*            hist      = (unsigned*)(ws + WS_HIST);
  unsigned*            counters  = (unsigned*)(ws + WS_COUNTERS);
  unsigned*            thr       = (unsigned*)(ws + WS_THR);
  unsigned long long*  cand      = (unsigned long long*)(ws + WS_CAND);
  unsigned*            top_idx   = (unsigned*)(ws + WS_TOPIDX);
  float*               top_score = (float*)(ws + WS_TOPSCORE);

  init_kernel<<<(3 * HIST_BINS + 255) / 256, 256, 0, stream>>>(hist, counters);

  for (int l = 0; l < 3; ++l) {
    int E4 = (A_ANCH * C_CLS * HWs[l]) / 4;
    int nchunks = E4 / 256;                        // exact: E4 % 256 == 0 for all levels
    int hblocks = nchunks < 1024 ? nchunks : 1024;
    hist_kernel<<<hblocks, 256, 0, stream>>>(obj[l], cls[l], HWs[l], hist + l * HIST_BINS, nchunks);
  }

  thresh_kernel<<<3, 256, 0, stream>>>(hist, thr);

  for (int l = 0; l < 3; ++l) {
    int E4 = (A_ANCH * C_CLS * HWs[l]) / 4;
    int cblocks = (E4 + 255) / 256; if (cblocks > 2048) cblocks = 2048;
    collect_kernel<<<cblocks, 256, 0, stream>>>(obj[l], cls[l], HWs[l], thr, l,
                                                counters + l, cand + (size_t)l * CAND_CAP, E4);
  }

  sortlevel_kernel<<<3, 256, 0, stream>>>(counters, cand, top_idx, top_score);

  final_kernel<<<1, 512, 0, stream>>>(top_idx, top_score, reg[0], reg[1], reg[2], (float*)d_out);
}